// GNNL_VP_54228257079467
// MI455X (gfx1250) — compile-verified
//
#include <hip/hip_runtime.h>
#include <hip/hip_bf16.h>

#define N_NODES 50000
#define E_EDGES 800000
#define F_IN_D  64
#define HID_D   128
#define LAT_D   128
#define DH_D    1024
#define G_D     256
#define EPS_BN  1e-5f

typedef float v2f __attribute__((ext_vector_type(2)));
typedef float v8f __attribute__((ext_vector_type(8)));

// ---------------------------------------------------------------------------
// Node-feature GEMM with register-resident B:  C[M,128] = A[M,K] * B[K,128]
// K is a compile-time constant (64 or 128), Nc = 128 fixed.
// block = 256 threads (8 waves); wave w owns output cols w*16..w*16+15.
// Each wave preloads its whole B operand (K/4 v2f fragments = K/2 VGPRs),
// then the block loops over row tiles: stage A tile (float4 -> ds b128),
// then K/4 back-to-back V_WMMA_F32_16X16X4_F32 with only LDS reads inside.
// ---------------------------------------------------------------------------
template <int K>
__global__ __launch_bounds__(256)
void gemm_wmma_f32_regB(const float* __restrict__ A, const float* __restrict__ B,
                        float* __restrict__ C, int M, int tilesPerBlock) {
    __shared__ float As[16][132];   // stride 132: b128 stores 16B-aligned,
                                    // frag reads hit banks (4m+k)%64 -> all 64
    const int tid    = threadIdx.x;
    const int wave   = tid >> 5;
    const int lane   = tid & 31;
    const int laneHi = lane >> 4;
    const int mfrag  = lane & 15;
    const int col    = wave * 16 + (lane & 15);

    // preload B fragments: frag i covers k = 4*i + laneHi*2 + {0,1}
    v2f bf[K / 4];
#pragma unroll
    for (int i = 0; i < K / 4; ++i) {
        const int k = 4 * i + laneHi * 2;
        bf[i].x = B[(size_t)k * 128 + col];
        bf[i].y = B[(size_t)(k + 1) * 128 + col];
    }

    for (int t = 0; t < tilesPerBlock; ++t) {
        const int rowBase = (blockIdx.y * tilesPerBlock + t) * 16;
        if (rowBase >= M) break;                 // uniform across block
        __syncthreads();                          // protect As from prev tile
        // stage A tile [16 x K] via float4 (rows are 16B-aligned: K%4==0)
        for (int i = tid; i < 4 * K; i += 256) {  // 4*K float4 total
            const int r  = i / (K / 4);
            const int c4 = i % (K / 4);
            float4 v = ((const float4*)(A + (size_t)(rowBase + r) * K))[c4];
            *(float4*)&As[r][c4 * 4] = v;
        }
        __syncthreads();

        v8f acc = {0.f, 0.f, 0.f, 0.f, 0.f, 0.f, 0.f, 0.f};
#pragma unroll
        for (int i = 0; i < K / 4; ++i) {
            const int kA = 4 * i + laneHi * 2;
            v2f a;
            a.x = As[mfrag][kA + 0];
            a.y = As[mfrag][kA + 1];
            acc = __builtin_amdgcn_wmma_f32_16x16x4_f32(
                false, a, false, bf[i], (short)0, acc, false, false);
        }
#pragma unroll
        for (int v = 0; v < 8; ++v) {
            const int m = v + laneHi * 8;         // C layout: vgpr v -> M=v+8*hi
            C[(size_t)(rowBase + m) * 128 + col] = acc[v];
        }
    }
}

// ---------------------------------------------------------------------------
// Generic fp32 WMMA GEMM (dense head):  C[M,Nc] = A[M,K] * B[K,Nc] (+bias/relu)
// grid = (Nc/128, M/16), block = 256 (8 waves). K arbitrary (scalar tail).
// ---------------------------------------------------------------------------
__global__ __launch_bounds__(256)
void gemm_wmma_f32(const float* __restrict__ A, const float* __restrict__ B,
                   float* __restrict__ C, const float* __restrict__ bias,
                   int M, int K, int Nc, int doRelu) {
    __shared__ float As[16][132];
    const int tid    = threadIdx.x;
    const int wave   = tid >> 5;
    const int lane   = tid & 31;
    const int laneHi = lane >> 4;
    const int mfrag  = lane & 15;
    const int rowBase = blockIdx.y * 16;
    const int col     = blockIdx.x * 128 + wave * 16 + (lane & 15);

    v8f acc = {0.f, 0.f, 0.f, 0.f, 0.f, 0.f, 0.f, 0.f};

    for (int k0 = 0; k0 < K; k0 += 128) {
        const int kcnt = (K - k0 < 128) ? (K - k0) : 128;
        if (kcnt == 128 && (K & 3) == 0) {
            // vectorized A staging: 512 float4, 2 per thread
            for (int i = tid; i < 512; i += 256) {
                const int r  = i >> 5;
                const int c4 = i & 31;
                float4 v = ((const float4*)(A + (size_t)(rowBase + r) * K + k0))[c4];
                *(float4*)&As[r][c4 * 4] = v;
            }
        } else {
            for (int i = tid; i < 16 * 128; i += 256) {
                const int r = i >> 7;
                const int c = i & 127;
                As[r][c] = (c < kcnt) ? A[(size_t)(rowBase + r) * K + (k0 + c)] : 0.f;
            }
        }
        __syncthreads();

        if (kcnt == 128) {
#pragma unroll 8
            for (int kk = 0; kk < 128; kk += 4) {
                const int kA = kk + laneHi * 2;
                v2f a, b;
                a.x = As[mfrag][kA + 0];
                a.y = As[mfrag][kA + 1];
                b.x = B[(size_t)(k0 + kA + 0) * Nc + col];
                b.y = B[(size_t)(k0 + kA + 1) * Nc + col];
                acc = __builtin_amdgcn_wmma_f32_16x16x4_f32(
                    false, a, false, b, (short)0, acc, false, false);
            }
        } else {
            int kk = 0;
            for (; kk + 4 <= kcnt; kk += 4) {
                const int kA = kk + laneHi * 2;
                v2f a, b;
                a.x = As[mfrag][kA + 0];
                a.y = As[mfrag][kA + 1];
                b.x = B[(size_t)(k0 + kA + 0) * Nc + col];
                b.y = B[(size_t)(k0 + kA + 1) * Nc + col];
                acc = __builtin_amdgcn_wmma_f32_16x16x4_f32(
                    false, a, false, b, (short)0, acc, false, false);
            }
            for (; kk < kcnt; ++kk) {             // scalar K tail (e.g. K=134)
                const float bv = B[(size_t)(k0 + kk) * Nc + col];
#pragma unroll
                for (int v = 0; v < 8; ++v)
                    acc[v] += As[v + laneHi * 8][kk] * bv;
            }
        }
        __syncthreads();
    }

    const float bv = bias ? bias[col] : 0.f;
#pragma unroll
    for (int v = 0; v < 8; ++v) {
        const int m   = v + laneHi * 8;
        const int row = rowBase + m;
        float val = acc[v] + bv;
        if (doRelu) val = fmaxf(val, 0.f);
        if (row < M) C[(size_t)row * Nc + col] = val;
    }
}

// ---------------------------------------------------------------------------
// Small utility / graph kernels
// ---------------------------------------------------------------------------
__global__ void fill_f32(float* p, float v, int n) {
    int i = blockIdx.x * blockDim.x + threadIdx.x;
    if (i < n) p[i] = v;
}

__global__ void deg_count(const int* __restrict__ dst, float* deg, int e_total) {
    int e = blockIdx.x * blockDim.x + threadIdx.x;
    if (e < e_total) atomicAdd(&deg[dst[e]], 1.0f);
}

__global__ void rsqrt_inplace(float* p, int n) {
    int i = blockIdx.x * blockDim.x + threadIdx.x;
    if (i < n) p[i] = rsqrtf(p[i]);
}

// agg[i][c] = xw[i][c] * dis[i]^2 + bias[c]   (self-loop term + bias)
__global__ void agg_init(const float* __restrict__ xw, const float* __restrict__ dis,
                         const float* __restrict__ bias, float* __restrict__ agg) {
    int i = blockIdx.x;
    int c = threadIdx.x;          // blockDim = 128
    float d = dis[i];
    agg[(size_t)i * 128 + c] = xw[(size_t)i * 128 + c] * d * d + bias[c];
}

// per-edge: agg[dst] += xw[src] * dis[src]*dis[dst]   (2 edges / 256-thr block)
__global__ void edge_scatter(const int* __restrict__ src, const int* __restrict__ dst,
                             const float* __restrict__ dis, const float* __restrict__ xw,
                             float* __restrict__ agg, int e_total) {
    int e = blockIdx.x * 2 + (threadIdx.x >> 7);
    int c = threadIdx.x & 127;
    if (e >= e_total) return;
    int s = src[e], d = dst[e];
    float nrm = dis[s] * dis[d];
    atomicAdd(&agg[(size_t)d * 128 + c], xw[(size_t)s * 128 + c] * nrm);
}

// column sum / sumsq over N rows -> stats[0..127]=sum, [128..255]=sumsq
__global__ void bn_stats(const float* __restrict__ h, float* stats, int n) {
    int c = threadIdx.x;          // blockDim = 128
    float s = 0.f, s2 = 0.f;
    for (int i = blockIdx.x; i < n; i += gridDim.x) {
        float v = h[(size_t)i * 128 + c];
        s += v; s2 += v * v;
    }
    atomicAdd(&stats[c], s);
    atomicAdd(&stats[128 + c], s2);
}

__global__ void bn_finalize(float* stats, const float* __restrict__ gamma,
                            const float* __restrict__ beta, float inv_n) {
    int c = threadIdx.x;          // single block of 128
    float mean = stats[c] * inv_n;
    float var  = stats[128 + c] * inv_n - mean * mean;
    float sc   = gamma[c] * rsqrtf(var + EPS_BN);
    stats[256 + c] = sc;
    stats[384 + c] = beta[c] - mean * sc;
}

__global__ void bn_apply(float* __restrict__ h, const float* __restrict__ stats,
                         int doRelu) {
    int i = blockIdx.x;
    int c = threadIdx.x;          // blockDim = 128
    float v = h[(size_t)i * 128 + c] * stats[256 + c] + stats[384 + c];
    if (doRelu) v = fmaxf(v, 0.f);
    h[(size_t)i * 128 + c] = v;
}

__global__ void pool_sum(const float* __restrict__ h, const int* __restrict__ batch,
                         float* __restrict__ psum, float* __restrict__ pcnt) {
    int i = blockIdx.x;
    int c = threadIdx.x;          // blockDim = 128
    int b = batch[i];
    atomicAdd(&psum[(size_t)b * 128 + c], h[(size_t)i * 128 + c]);
    if (c == 0) atomicAdd(&pcnt[b], 1.0f);
}

// z[g] = [ pooled_mean(g) (128) | origin(g) (3) | direction(g) (3) ] -> [G,134]
__global__ void build_z(const float* __restrict__ psum, const float* __restrict__ pcnt,
                        const float* __restrict__ origin, const float* __restrict__ dir,
                        float* __restrict__ z) {
    int g = blockIdx.x;
    int c = threadIdx.x;          // blockDim = 134
    float v;
    if (c < 128)       v = psum[(size_t)g * 128 + c] / fmaxf(pcnt[g], 1.0f);
    else if (c < 131)  v = origin[g * 3 + (c - 128)];
    else               v = dir[g * 3 + (c - 131)];
    z[(size_t)g * 134 + c] = v;
}

// final [G,1024] x [1024,3] + bout : one block per graph, tree-reduce
__global__ void out_head(const float* __restrict__ zh, const float* __restrict__ Wout,
                         const float* __restrict__ bout, float* __restrict__ out) {
    __shared__ float red[128][3];
    int g = blockIdx.x;
    int t = threadIdx.x;          // blockDim = 128
    float a0 = 0.f, a1 = 0.f, a2 = 0.f;
    for (int k = t; k < DH_D; k += 128) {
        float v = zh[(size_t)g * DH_D + k];
        a0 += v * Wout[k * 3 + 0];
        a1 += v * Wout[k * 3 + 1];
        a2 += v * Wout[k * 3 + 2];
    }
    red[t][0] = a0; red[t][1] = a1; red[t][2] = a2;
    __syncthreads();
    for (int s = 64; s > 0; s >>= 1) {
        if (t < s) {
            red[t][0] += red[t + s][0];
            red[t][1] += red[t + s][1];
            red[t][2] += red[t + s][2];
        }
        __syncthreads();
    }
    if (t < 3) out[g * 3 + t] = red[0][t] + bout[t];
}

// ---------------------------------------------------------------------------
extern "C" void kernel_launch(void* const* d_in, const int* in_sizes, int n_in,
                              void* d_out, int out_size, void* d_ws, size_t ws_size,
                              hipStream_t stream) {
    (void)in_sizes; (void)n_in; (void)out_size; (void)ws_size;
    const float* x      = (const float*)d_in[0];
    const float* origin = (const float*)d_in[1];
    const float* dirn   = (const float*)d_in[2];
    const int*   ei     = (const int*)d_in[3];     // [2, E]
    const int*   batch  = (const int*)d_in[4];
    const float* W1  = (const float*)d_in[5];  const float* b1  = (const float*)d_in[6];
    const float* W2  = (const float*)d_in[7];  const float* b2  = (const float*)d_in[8];
    const float* W3  = (const float*)d_in[9];  const float* b3  = (const float*)d_in[10];
    const float* g1  = (const float*)d_in[11]; const float* be1 = (const float*)d_in[12];
    const float* g2  = (const float*)d_in[13]; const float* be2 = (const float*)d_in[14];
    const float* g3  = (const float*)d_in[15]; const float* be3 = (const float*)d_in[16];
    const float* Wd0 = (const float*)d_in[17]; const float* bd0 = (const float*)d_in[18];
    const float* Wd1 = (const float*)d_in[19]; const float* bd1 = (const float*)d_in[20];
    const float* Wd2 = (const float*)d_in[21]; const float* bd2 = (const float*)d_in[22];
    const float* Wout = (const float*)d_in[23]; const float* bout = (const float*)d_in[24];
    float* out = (float*)d_out;

    const int* src = ei;
    const int* dst = ei + E_EDGES;

    // workspace layout
    float* ws    = (float*)d_ws;
    float* dis   = ws;                              // N
    float* buf1  = dis  + N_NODES;                  // N*128  (xW)
    float* buf2  = buf1 + (size_t)N_NODES * 128;    // N*128  (agg / h)
    float* stats = buf2 + (size_t)N_NODES * 128;    // 512
    float* psum  = stats + 512;                     // G*128
    float* pcnt  = psum + (size_t)G_D * 128;        // G
    float* zbuf  = pcnt + G_D;                      // G*134
    float* zh1   = zbuf + (size_t)G_D * 134;        // G*1024
    float* zh2   = zh1  + (size_t)G_D * 1024;       // G*1024

    const float invN = 1.0f / (float)N_NODES;

    // --- degrees (self-loop +1) and D^{-1/2} ---
    fill_f32<<<(N_NODES + 255) / 256, 256, 0, stream>>>(dis, 1.0f, N_NODES);
    deg_count<<<(E_EDGES + 255) / 256, 256, 0, stream>>>(dst, dis, E_EDGES);
    rsqrt_inplace<<<(N_NODES + 255) / 256, 256, 0, stream>>>(dis, N_NODES);

    const dim3 blkG(256);
    const int  TPB = 5;                              // row tiles per block
    const dim3 grdReg(1, (N_NODES / 16 + TPB - 1) / TPB);   // 625 blocks

    // --- GCN layer 1 ---
    gemm_wmma_f32_regB<F_IN_D><<<grdReg, blkG, 0, stream>>>(x, W1, buf1,
                                                            N_NODES, TPB);
    agg_init<<<N_NODES, 128, 0, stream>>>(buf1, dis, b1, buf2);
    edge_scatter<<<E_EDGES / 2, 256, 0, stream>>>(src, dst, dis, buf1, buf2, E_EDGES);
    fill_f32<<<2, 128, 0, stream>>>(stats, 0.0f, 256);
    bn_stats<<<512, 128, 0, stream>>>(buf2, stats, N_NODES);
    bn_finalize<<<1, 128, 0, stream>>>(stats, g1, be1, invN);
    bn_apply<<<N_NODES, 128, 0, stream>>>(buf2, stats, 1);

    // --- GCN layer 2 ---
    gemm_wmma_f32_regB<HID_D><<<grdReg, blkG, 0, stream>>>(buf2, W2, buf1,
                                                           N_NODES, TPB);
    agg_init<<<N_NODES, 128, 0, stream>>>(buf1, dis, b2, buf2);
    edge_scatter<<<E_EDGES / 2, 256, 0, stream>>>(src, dst, dis, buf1, buf2, E_EDGES);
    fill_f32<<<2, 128, 0, stream>>>(stats, 0.0f, 256);
    bn_stats<<<512, 128, 0, stream>>>(buf2, stats, N_NODES);
    bn_finalize<<<1, 128, 0, stream>>>(stats, g2, be2, invN);
    bn_apply<<<N_NODES, 128, 0, stream>>>(buf2, stats, 1);

    // --- GCN layer 3 (no relu) ---
    gemm_wmma_f32_regB<HID_D><<<grdReg, blkG, 0, stream>>>(buf2, W3, buf1,
                                                           N_NODES, TPB);
    agg_init<<<N_NODES, 128, 0, stream>>>(buf1, dis, b3, buf2);
    edge_scatter<<<E_EDGES / 2, 256, 0, stream>>>(src, dst, dis, buf1, buf2, E_EDGES);
    fill_f32<<<2, 128, 0, stream>>>(stats, 0.0f, 256);
    bn_stats<<<512, 128, 0, stream>>>(buf2, stats, N_NODES);
    bn_finalize<<<1, 128, 0, stream>>>(stats, g3, be3, invN);
    bn_apply<<<N_NODES, 128, 0, stream>>>(buf2, stats, 0);

    // --- global mean pool + concat ---
    fill_f32<<<(G_D * 128 + G_D + 255) / 256, 256, 0, stream>>>(psum, 0.0f,
                                                                G_D * 128 + G_D);
    pool_sum<<<N_NODES, 128, 0, stream>>>(buf2, batch, psum, pcnt);
    build_z<<<G_D, 134, 0, stream>>>(psum, pcnt, origin, dirn, zbuf);

    // --- dense head (WMMA GEMMs with fused bias+relu) ---
    gemm_wmma_f32<<<dim3(DH_D / 128, G_D / 16), blkG, 0, stream>>>(
        zbuf, Wd0, zh1, bd0, G_D, LAT_D + 6, DH_D, 1);
    gemm_wmma_f32<<<dim3(DH_D / 128, G_D / 16), blkG, 0, stream>>>(
        zh1, Wd1, zh2, bd1, G_D, DH_D, DH_D, 1);
    gemm_wmma_f32<<<dim3(DH_D / 128, G_D / 16), blkG, 0, stream>>>(
        zh2, Wd2, zh1, bd2, G_D, DH_D, DH_D, 1);
    out_head<<<G_D, 128, 0, stream>>>(zh1, Wout, bout, out);
}